// CosimLoss_4020089389403
// MI455X (gfx1250) — compile-verified
//
#include <hip/hip_runtime.h>
#include <math.h>
#include <stdint.h>

typedef float v2f __attribute__((ext_vector_type(2)));
typedef float v8f __attribute__((ext_vector_type(8)));
typedef unsigned int u32;
typedef u32 u32x4 __attribute__((ext_vector_type(4)));
typedef u32 u32x8 __attribute__((ext_vector_type(8)));

#define B_    16
#define H_    512
#define W_    512
#define NW    16
#define OUT_  497        // H_ - NW + 1
#define TILES1D 32       // ceil(497/16)
#define IMGSZ (H_*W_)

// ---------------------------------------------------------------------------
// Kernel 1: bilinear grid-sample (padding_mode='zeros', align_corners=False)
// plus the three per-pixel products a*a, a*b, b*b.
// ---------------------------------------------------------------------------
__global__ __launch_bounds__(256)
void sample_products(const float* __restrict__ rep1, const float* __restrict__ rep2,
                     const float* __restrict__ grid,
                     float* __restrict__ AA, float* __restrict__ AB, float* __restrict__ BB)
{
    int idx = blockIdx.x * blockDim.x + threadIdx.x;   // over B*H*W
    if (idx >= B_ * IMGSZ) return;
    int b = idx / IMGSZ;

    float2 g2 = ((const float2*)grid)[idx];            // global_load_b64
    float ix = ((g2.x + 1.0f) * (float)W_ - 1.0f) * 0.5f;
    float iy = ((g2.y + 1.0f) * (float)H_ - 1.0f) * 0.5f;
    float x0f = floorf(ix), y0f = floorf(iy);
    int   x0  = (int)x0f,  y0  = (int)y0f;
    float wx1 = ix - x0f, wx0 = 1.0f - wx1;
    float wy1 = iy - y0f, wy0 = 1.0f - wy1;

    const float* img = rep2 + (size_t)b * IMGSZ;
    auto g = [&](int xi, int yi) -> float {
        bool inb = (xi >= 0) & (xi < W_) & (yi >= 0) & (yi < H_);
        int xc = min(max(xi, 0), W_ - 1);
        int yc = min(max(yi, 0), H_ - 1);
        float v = img[yc * W_ + xc];                   // clamped -> unconditional load
        return inb ? v : 0.0f;                         // zeros padding via cndmask
    };
    float bval = g(x0,     y0    ) * (wx0 * wy0)
               + g(x0 + 1, y0    ) * (wx1 * wy0)
               + g(x0,     y0 + 1) * (wx0 * wy1)
               + g(x0 + 1, y0 + 1) * (wx1 * wy1);
    float aval = rep1[idx];

    AA[idx] = aval * aval;
    AB[idx] = aval * bval;
    BB[idx] = bval * bval;
}

// ---------------------------------------------------------------------------
// Tensor Data Mover: DMA a 32x32 f32 tile (with zero-fill clipping at the
// image edge) from global memory into LDS.  D# built in SGPRs per
// CDNA5 ISA; 2-group form (VADDR2/3 = NULL -> 2D tensor).
//   group0: [1:0]=count=1, [63:32]=lds_addr, [120:64]=global_addr, [127:126]=type=2
//   group1: data_size=4B, tensor_dim0/1 = remaining extent from tile origin
//           (OOB reads return zero -> free edge clipping), tile 32x32, stride 512
// ---------------------------------------------------------------------------
__device__ __attribute__((always_inline))
void tdm_load_tile(u32 lds_off, const float* gsrc, int tdim0, int tdim1)
{
    unsigned long long ga = (unsigned long long)(uintptr_t)gsrc;
    u32x4 g0;
    g0[0] = 1u;                                            // count=1, is_restore=0
    g0[1] = lds_off;                                       // lds_addr (bytes)
    g0[2] = (u32)(ga & 0xFFFFFFFFu);                       // global_addr[31:0]
    g0[3] = (u32)((ga >> 32) & 0x01FFFFFFu) | 0x80000000u; // addr[56:32] | type=2

    u32x8 g1;
    g1[0] = 2u << 16;                                      // data_size=2 (4 bytes)
    g1[1] = ((u32)tdim0 & 0xFFFFu) << 16;                  // tensor_dim0[15:0]
    g1[2] = ((u32)tdim0 >> 16) | (((u32)tdim1 & 0xFFFFu) << 16); // dim0 hi | dim1 lo
    g1[3] = ((u32)tdim1 >> 16) | (32u << 16);              // dim1 hi | tile_dim0=32
    g1[4] = 32u;                                           // tile_dim1=32, tile_dim2=0
    g1[5] = (u32)W_;                                       // tensor_dim0_stride=512
    g1[6] = 0u;
    g1[7] = 0u;

    asm volatile("tensor_load_to_lds %0, %1"
                 :: "s"(g0), "s"(g1)
                 : "memory");
}

// ---------------------------------------------------------------------------
// 16x16 sliding-window box sum via V_WMMA_F32_16X16X4_F32.
//   Sliding sum == banded-ones matmul: OUT = Wv * X * Wh^T per 16x16 tile,
//   triangular-ones operands generated from lane id, data from the LDS tile.
// One wave (32 threads) per output tile; lin = 32x32 row-major LDS tile.
// ---------------------------------------------------------------------------
__device__ __attribute__((always_inline))
v8f win16_lds(const float* lin, float* ldsT, int lane)
{
    const int i    = lane & 15;    // A row / B col / D col (N)
    const int half = lane >> 4;    // 0 or 1

    // ---- vertical pass: T[cblk] = U * X(rows 0..15) + L * X(rows 16..31) ----
    for (int cblk = 0; cblk < 2; ++cblk) {
        v8f t = {0.f, 0.f, 0.f, 0.f, 0.f, 0.f, 0.f, 0.f};
        for (int kk = 0; kk < 4; ++kk) {
            const int kg = kk * 4 + half * 2;   // this lane's K pair
            v2f a, bb;
            // U (upper-tri incl diag): A[i][k] = (k >= i)
            a.x  = (kg     >= i) ? 1.0f : 0.0f;
            a.y  = (kg + 1 >= i) ? 1.0f : 0.0f;
            bb.x = lin[(kg    ) * 32 + cblk * 16 + i];
            bb.y = lin[(kg + 1) * 32 + cblk * 16 + i];
            t = __builtin_amdgcn_wmma_f32_16x16x4_f32(false, a, false, bb,
                                                      (short)0, t, false, false);
            // L (strict lower): A[i][k] = (k < i), rows 16..31
            a.x  = (kg     < i) ? 1.0f : 0.0f;
            a.y  = (kg + 1 < i) ? 1.0f : 0.0f;
            bb.x = lin[(16 + kg) * 32 + cblk * 16 + i];
            bb.y = lin[(17 + kg) * 32 + cblk * 16 + i];
            t = __builtin_amdgcn_wmma_f32_16x16x4_f32(false, a, false, bb,
                                                      (short)0, t, false, false);
        }
        // C/D layout -> LDS row-major: element v is (M = v + half*8, N = i)
        for (int v = 0; v < 8; ++v)
            ldsT[cblk * 256 + (v + half * 8) * 16 + i] = t[v];
    }
    __syncthreads();

    // ---- horizontal pass: OUT = T0 * L1 + T1 * L2 ----
    v8f out = {0.f, 0.f, 0.f, 0.f, 0.f, 0.f, 0.f, 0.f};
    for (int kk = 0; kk < 4; ++kk) {
        const int kg = kk * 4 + half * 2;
        v2f a, bb;
        // T0 as A;  L1[c][j] = (j <= c) as B (row c = kg, col j = i)
        a.x  = ldsT[0 * 256 + i * 16 + kg];
        a.y  = ldsT[0 * 256 + i * 16 + kg + 1];
        bb.x = (i <= kg    ) ? 1.0f : 0.0f;
        bb.y = (i <= kg + 1) ? 1.0f : 0.0f;
        out = __builtin_amdgcn_wmma_f32_16x16x4_f32(false, a, false, bb,
                                                    (short)0, out, false, false);
        // T1 as A;  L2[c][j] = (c < j) as B
        a.x  = ldsT[1 * 256 + i * 16 + kg];
        a.y  = ldsT[1 * 256 + i * 16 + kg + 1];
        bb.x = (kg     < i) ? 1.0f : 0.0f;
        bb.y = (kg + 1 < i) ? 1.0f : 0.0f;
        out = __builtin_amdgcn_wmma_f32_16x16x4_f32(false, a, false, bb,
                                                    (short)0, out, false, false);
    }
    __syncthreads();   // protect ldsT before next quantity reuses it
    return out;
}

__global__ __launch_bounds__(32)
void cosim_tiles(const float* __restrict__ AA, const float* __restrict__ AB,
                 const float* __restrict__ BB, float* __restrict__ partials)
{
    __shared__ float ldsA[1024];   // staged 32x32 input tiles (TDM destinations)
    __shared__ float ldsB[1024];
    __shared__ float ldsC[1024];
    __shared__ float ldsT[512];    // vertical-pass intermediate

    const int tile = blockIdx.x;           // B_ * 1024 tiles
    const int b  = tile >> 10;
    const int t2 = tile & 1023;
    const int tr = t2 >> 5;
    const int tc = t2 & 31;
    const int R = tr * 16, C = tc * 16;
    const int lane = threadIdx.x;

    const size_t base = (size_t)b * IMGSZ + (size_t)R * W_ + C;
    const int tdim0 = W_ - C;              // remaining columns from tile origin
    const int tdim1 = H_ - R;              // remaining rows

    // Issue all three DMA tile loads, then consume with staged waits so the
    // AA window-sum WMMAs overlap the AB/BB transfers (in-order completion).
    tdm_load_tile((u32)(uintptr_t)ldsA, AA + base, tdim0, tdim1);
    tdm_load_tile((u32)(uintptr_t)ldsB, AB + base, tdim0, tdim1);
    tdm_load_tile((u32)(uintptr_t)ldsC, BB + base, tdim0, tdim1);

    asm volatile("s_wait_tensorcnt 0x2" ::: "memory");
    v8f saa = win16_lds(ldsA, ldsT, lane);
    asm volatile("s_wait_tensorcnt 0x1" ::: "memory");
    v8f sab = win16_lds(ldsB, ldsT, lane);
    asm volatile("s_wait_tensorcnt 0x0" ::: "memory");
    v8f sbb = win16_lds(ldsC, ldsT, lane);

    const int i = lane & 15, half = lane >> 4;
    float partial = 0.0f;
    for (int v = 0; v < 8; ++v) {
        int orow = R + v + half * 8;       // D-matrix M
        int ocol = C + i;                  // D-matrix N
        if (orow < OUT_ && ocol < OUT_) {
            float na = fmaxf(sqrtf(saa[v]), 1e-12f);
            float nb = fmaxf(sqrtf(sbb[v]), 1e-12f);
            float cs = sab[v] / (na * nb);
            partial += ((cs != 0.0f) ? 1.0f : 0.0f) - cs;  // nonzero count - cosim
        }
    }
    for (int s = 16; s >= 1; s >>= 1)      // wave32 butterfly reduction
        partial += __shfl_xor(partial, s, 32);
    if (lane == 0) partials[tile] = partial;
}

// ---------------------------------------------------------------------------
// Kernel 3: deterministic tree reduction of per-tile partials -> scalar loss.
// ---------------------------------------------------------------------------
__global__ __launch_bounds__(1024)
void reduce_partials(const float* __restrict__ partials, int n, float* __restrict__ out)
{
    __shared__ float s[1024];
    float acc = 0.0f;
    for (int idx = threadIdx.x; idx < n; idx += 1024) acc += partials[idx];
    s[threadIdx.x] = acc;
    __syncthreads();
    for (int stride = 512; stride >= 1; stride >>= 1) {
        if ((int)threadIdx.x < stride) s[threadIdx.x] += s[threadIdx.x + stride];
        __syncthreads();
    }
    if (threadIdx.x == 0)
        out[0] = s[0] / ((float)B_ * (float)OUT_ * (float)OUT_);
}

// ---------------------------------------------------------------------------
extern "C" void kernel_launch(void* const* d_in, const int* in_sizes, int n_in,
                              void* d_out, int out_size, void* d_ws, size_t ws_size,
                              hipStream_t stream)
{
    const float* rep1 = (const float*)d_in[0];   // (16,1,512,512) f32
    const float* rep2 = (const float*)d_in[1];   // (16,1,512,512) f32
    const float* grid = (const float*)d_in[2];   // (16,512,512,2) f32

    float* ws = (float*)d_ws;
    float* AA = ws;
    float* AB = AA + (size_t)B_ * IMGSZ;
    float* BB = AB + (size_t)B_ * IMGSZ;
    float* partials = BB + (size_t)B_ * IMGSZ;   // 16384 floats

    const int npix   = B_ * IMGSZ;
    const int ntiles = B_ * TILES1D * TILES1D;

    sample_products<<<(npix + 255) / 256, 256, 0, stream>>>(rep1, rep2, grid, AA, AB, BB);
    cosim_tiles<<<ntiles, 32, 0, stream>>>(AA, AB, BB, partials);
    reduce_partials<<<1, 1024, 0, stream>>>(partials, ntiles, (float*)d_out);
}